// Block_9397388444369
// MI455X (gfx1250) — compile-verified
//
#include <hip/hip_runtime.h>

// Problem constants (B=2, T=512, C=256, HID=1024)
#define NROW 1024          // B*T
#define CDIM 256
#define HIDD 1024
#define EPSV 1e-5f

typedef float v2f __attribute__((ext_vector_type(2)));
typedef float v8f __attribute__((ext_vector_type(8)));

// ---------------------------------------------------------------------------
// Kernel 1: x2 = x + bp ; h2 = LN(x2)*g2 + b2   (one wave32 per row)
// ---------------------------------------------------------------------------
__global__ void ln_kernel(const float* __restrict__ x, const float* __restrict__ bp,
                          const float* __restrict__ g2, const float* __restrict__ b2,
                          float* __restrict__ x2, float* __restrict__ h2) {
    const int row  = blockIdx.x * 8 + (threadIdx.x >> 5);
    const int lane = threadIdx.x & 31;
    const float* xr  = x  + row * CDIM;
    float*       x2r = x2 + row * CDIM;

    float v[8];
    float s = 0.f;
#pragma unroll
    for (int i = 0; i < 8; ++i) {
        const int c = lane + i * 32;
        v[i] = xr[c] + bp[c];
        x2r[c] = v[i];
        s += v[i];
    }
#pragma unroll
    for (int off = 16; off > 0; off >>= 1) s += __shfl_xor(s, off, 32);
    const float mu = s * (1.f / CDIM);

    float q = 0.f;
#pragma unroll
    for (int i = 0; i < 8; ++i) { const float d = v[i] - mu; q += d * d; }
#pragma unroll
    for (int off = 16; off > 0; off >>= 1) q += __shfl_xor(q, off, 32);
    const float rstd = rsqrtf(q * (1.f / CDIM) + EPSV);

    float* h2r = h2 + row * CDIM;
#pragma unroll
    for (int i = 0; i < 8; ++i) {
        const int c = lane + i * 32;
        h2r[c] = (v[i] - mu) * rstd * g2[c] + b2[c];
    }
}

// ---------------------------------------------------------------------------
// Kernel 2: act = relu(h2 @ W1^T + bf1)   (NROW x HIDD), K = CDIM
// Each wave owns a 16x64 strip: A-frag reused across 4 independent WMMA
// accumulators (1.25 loads/WMMA, 4-way XDL chain interleave).
// ---------------------------------------------------------------------------
__global__ void gemm1_kernel(const float* __restrict__ h2, const float* __restrict__ W1,
                             const float* __restrict__ bf1, float* __restrict__ act) {
    const int lane = threadIdx.x & 31;
    const int wave = threadIdx.x >> 5;
    const int m0   = blockIdx.y * 16;
    const int n0   = (blockIdx.x * 4 + wave) * 64;   // 64-col strip
    const int half = lane >> 4;        // 0: K={0,1}, 1: K={2,3}
    const int l15  = lane & 15;
    const int koff = half * 2;

    const float* Ap = h2 + (size_t)(m0 + l15) * CDIM + koff;
    const float* Bp = W1 + (size_t)(n0 + l15) * CDIM + koff;

    v8f acc0 = {}, acc1 = {}, acc2 = {}, acc3 = {};
#pragma unroll 2
    for (int k = 0; k < CDIM; k += 4) {
        v2f a  = *(const v2f*)(Ap + k);
        v2f b0 = *(const v2f*)(Bp + k);
        v2f b1 = *(const v2f*)(Bp + 16 * CDIM + k);
        v2f b2 = *(const v2f*)(Bp + 32 * CDIM + k);
        v2f b3 = *(const v2f*)(Bp + 48 * CDIM + k);
        acc0 = __builtin_amdgcn_wmma_f32_16x16x4_f32(false, a, false, b0, (short)0, acc0, false, false);
        acc1 = __builtin_amdgcn_wmma_f32_16x16x4_f32(false, a, false, b1, (short)0, acc1, false, false);
        acc2 = __builtin_amdgcn_wmma_f32_16x16x4_f32(false, a, false, b2, (short)0, acc2, false, false);
        acc3 = __builtin_amdgcn_wmma_f32_16x16x4_f32(false, a, false, b3, (short)0, acc3, false, false);
    }

#define G1_EPI(ACC, T)                                                        \
    {                                                                         \
        const int n = n0 + (T) * 16 + l15;                                    \
        const float bias = bf1[n];                                            \
        _Pragma("unroll")                                                     \
        for (int r = 0; r < 8; ++r) {                                         \
            const int m = m0 + r + half * 8;                                  \
            float val = ACC[r] + bias;                                        \
            act[(size_t)m * HIDD + n] = val > 0.f ? val : 0.f;                \
        }                                                                     \
    }
    G1_EPI(acc0, 0) G1_EPI(acc1, 1) G1_EPI(acc2, 2) G1_EPI(acc3, 3)
#undef G1_EPI
}

// ---------------------------------------------------------------------------
// Kernel 3: out = x2 + act @ W2^T + bf2   (NROW x CDIM), K = HIDD
// Same 16x64-strip-per-wave structure.
// ---------------------------------------------------------------------------
__global__ void gemm2_kernel(const float* __restrict__ act, const float* __restrict__ W2,
                             const float* __restrict__ bf2, const float* __restrict__ x2,
                             float* __restrict__ out) {
    const int lane = threadIdx.x & 31;
    const int wave = threadIdx.x >> 5;
    const int m0   = blockIdx.y * 16;
    const int n0   = wave * 64;                      // grid.x == 1, 4 waves span 256 cols
    const int half = lane >> 4;
    const int l15  = lane & 15;
    const int koff = half * 2;

    const float* Ap = act + (size_t)(m0 + l15) * HIDD + koff;
    const float* Bp = W2  + (size_t)(n0 + l15) * HIDD + koff;

    v8f acc0 = {}, acc1 = {}, acc2 = {}, acc3 = {};
#pragma unroll 2
    for (int k = 0; k < HIDD; k += 4) {
        v2f a  = *(const v2f*)(Ap + k);
        v2f b0 = *(const v2f*)(Bp + k);
        v2f b1 = *(const v2f*)(Bp + 16 * HIDD + k);
        v2f b2 = *(const v2f*)(Bp + 32 * HIDD + k);
        v2f b3 = *(const v2f*)(Bp + 48 * HIDD + k);
        acc0 = __builtin_amdgcn_wmma_f32_16x16x4_f32(false, a, false, b0, (short)0, acc0, false, false);
        acc1 = __builtin_amdgcn_wmma_f32_16x16x4_f32(false, a, false, b1, (short)0, acc1, false, false);
        acc2 = __builtin_amdgcn_wmma_f32_16x16x4_f32(false, a, false, b2, (short)0, acc2, false, false);
        acc3 = __builtin_amdgcn_wmma_f32_16x16x4_f32(false, a, false, b3, (short)0, acc3, false, false);
    }

#define G2_EPI(ACC, T)                                                        \
    {                                                                         \
        const int n = n0 + (T) * 16 + l15;                                    \
        const float bias = bf2[n];                                            \
        _Pragma("unroll")                                                     \
        for (int r = 0; r < 8; ++r) {                                         \
            const int m = m0 + r + half * 8;                                  \
            out[(size_t)m * CDIM + n] =                                       \
                x2[(size_t)m * CDIM + n] + ACC[r] + bias;                     \
        }                                                                     \
    }
    G2_EPI(acc0, 0) G2_EPI(acc1, 1) G2_EPI(acc2, 2) G2_EPI(acc3, 3)
#undef G2_EPI
}

// ---------------------------------------------------------------------------
extern "C" void kernel_launch(void* const* d_in, const int* in_sizes, int n_in,
                              void* d_out, int out_size, void* d_ws, size_t ws_size,
                              hipStream_t stream) {
    // setup_inputs order: x, Wt, Wp, bp, g1, b1, g2, b2, W1, bf1, W2, bf2
    const float* x   = (const float*)d_in[0];
    const float* bp  = (const float*)d_in[3];
    const float* g2  = (const float*)d_in[6];
    const float* b2  = (const float*)d_in[7];
    const float* W1  = (const float*)d_in[8];
    const float* bf1 = (const float*)d_in[9];
    const float* W2  = (const float*)d_in[10];
    const float* bf2 = (const float*)d_in[11];
    float* out = (float*)d_out;

    // workspace layout: x2 (1 MB) | h2 (1 MB) | act (4 MB)
    float* x2  = (float*)d_ws;
    float* h2  = x2 + (size_t)NROW * CDIM;
    float* act = h2 + (size_t)NROW * CDIM;

    // 1) residual bias-add + LayerNorm: 128 blocks x 256 threads (8 rows/block)
    ln_kernel<<<dim3(NROW / 8), dim3(256), 0, stream>>>(x, bp, g2, b2, x2, h2);

    // 2) GEMM1 + ReLU: 16x64 strip per wave, 4 waves/block
    //    grid = (HIDD/256 strip-groups, NROW/16 m-tiles)
    gemm1_kernel<<<dim3(HIDD / 256, NROW / 16), dim3(128), 0, stream>>>(h2, W1, bf1, act);

    // 3) GEMM2 + residual: 4 waves cover all 256 cols; grid = (1, NROW/16)
    gemm2_kernel<<<dim3(1, NROW / 16), dim3(128), 0, stream>>>(act, W2, bf2, x2, out);
}